// ProbeDirectionRemovalModule_7481833029919
// MI455X (gfx1250) — compile-verified
//
#include <hip/hip_runtime.h>
#include <math.h>

typedef float v2f __attribute__((ext_vector_type(2)));
typedef float v8f __attribute__((ext_vector_type(8)));

#define H_DIM 1024
#define QSTRIDE 18               // padded LDS row stride (floats): 18*m mod 64 distinct for m=0..15
#define WAVES_PER_BLOCK 8
#define TOK_PER_TILE 16

#define SMEM_Q_FLOATS   (H_DIM * QSTRIDE)                         // 18432
#define SMEM_COEF_FLOATS (WAVES_PER_BLOCK * TOK_PER_TILE * QSTRIDE) // 2304
#define SMEM_BYTES ((SMEM_Q_FLOATS + SMEM_COEF_FLOATS) * sizeof(float))

__global__ __launch_bounds__(256) void probe_removal_kernel(
    const float* __restrict__ hs, const float* __restrict__ Q,
    float* __restrict__ out, int nTokens)
{
  extern __shared__ float smem[];
  float* lds_q    = smem;                  // Q[j][i] padded: lds_q[j*18 + i]
  float* lds_coef = smem + SMEM_Q_FLOATS;  // per-wave 16x16 (padded) A-matrix for pass 2

  const int tid  = threadIdx.x;
  const int wave = tid >> 5;
  const int lane = tid & 31;
  const int m    = lane & 15;   // M (pass1-A) / N (B) lane index
  const int g    = lane >> 4;   // lane half selects K pair / row offset +8

  // ---- cooperative load of Q [1024 x 16] into padded LDS (coalesced b128) ----
  {
    const float4* q4 = reinterpret_cast<const float4*>(Q);   // 4096 float4 total
    #pragma unroll
    for (int it = 0; it < 16; ++it) {
      int linear = tid + it * 256;
      int j = linear >> 2;            // hidden row
      int c = (linear & 3) << 2;      // col group within row
      float4 v = q4[linear];
      float* dst = &lds_q[j * QSTRIDE + c];
      dst[0] = v.x; dst[1] = v.y; dst[2] = v.z; dst[3] = v.w;
    }
  }
  __syncthreads();

  const int tile    = blockIdx.x * WAVES_PER_BLOCK + wave;
  const int tokBase = tile * TOK_PER_TILE;
  if (tokBase >= nTokens) return;      // wave-uniform: EXEC stays all-1s for WMMA

  // ================= Pass 1: C = Htile[16x1024] * Q[1024x16] =================
  // f32 WMMA A layout: lane m, VGPR v -> element (m, k0 + 2g + v); one b64/lane/iter
  const float* rowp = hs + (size_t)(tokBase + m) * H_DIM + (g << 1);
  v8f   coef = {};     // C/D layout: (lane,vgpr v) -> row v+8g, col m
  float old2 = 0.0f;   // partial ||h||^2 of row m over this half's columns

  #pragma unroll 8
  for (int k0 = 0; k0 < H_DIM; k0 += 4) {
    v2f a = *reinterpret_cast<const v2f*>(rowp + k0);
    const float* qp = &lds_q[(k0 + (g << 1)) * QSTRIDE + m];
    v2f b;
    b.x = qp[0];
    b.y = qp[QSTRIDE];
    coef = __builtin_amdgcn_wmma_f32_16x16x4_f32(false, a, false, b,
                                                 (short)0, coef, false, false);
    old2 = fmaf(a.x, a.x, old2);
    old2 = fmaf(a.y, a.y, old2);
  }
  old2 += __shfl_xor(old2, 16, 32);   // other half of row m's columns

  // ||c||^2 per row (row = v + 8g): reduce coef[v]^2 across the 16 lanes of a half
  float sc[8];
  #pragma unroll
  for (int v = 0; v < 8; ++v) {
    float s = coef[v] * coef[v];
    s += __shfl_xor(s, 1, 32);
    s += __shfl_xor(s, 2, 32);
    s += __shfl_xor(s, 4, 32);
    s += __shfl_xor(s, 8, 32);
    sc[v] = s;
  }

  // analytic rescale: s_row = ||h|| / ||h'|| = sqrt(old2 / (old2 - ||c||^2))
  float srow[8];
  #pragma unroll
  for (int v = 0; v < 8; ++v) {
    float o2 = __shfl(old2, (g << 3) + v, 32);   // old2 of row v+8g lives in lane (v+8g)
    float denom = fmaxf(o2 - sc[v], 1e-30f);
    srow[v] = sqrtf(o2 / denom);
  }

  // relayout C/D -> A through LDS, folding negate + scale: A2 = -(s_row * coef)
  float* cw = &lds_coef[wave * TOK_PER_TILE * QSTRIDE];
  #pragma unroll
  for (int v = 0; v < 8; ++v)
    cw[(v + (g << 3)) * QSTRIDE + m] = -(coef[v] * srow[v]);
  asm volatile("s_wait_dscnt 0" ::: "memory");   // same-wave DS RAW before fragment reads

  // pass-2 A fragments (loop invariant): lane m, k = 4c + 2g + {0,1}
  v2f af[4];
  #pragma unroll
  for (int c = 0; c < 4; ++c)
    af[c] = *reinterpret_cast<const v2f*>(&cw[m * QSTRIDE + 4 * c + (g << 1)]);

  // ======== Pass 2: D = s*H + (-s*C) * Q^T over 64 column blocks of 16 ========
  const float* hacc = hs  + (size_t)(tokBase + (g << 3)) * H_DIM + m;
  float*       oacc = out + (size_t)(tokBase + (g << 3)) * H_DIM + m;

  for (int t = 0; t < H_DIM / 16; ++t) {
    v8f d;
    #pragma unroll
    for (int v = 0; v < 8; ++v)                       // acc init = s_row * H (C/D layout)
      d[v] = hacc[(size_t)v * H_DIM + t * 16] * srow[v];
    #pragma unroll
    for (int c = 0; c < 4; ++c) {                     // B = Q^T block: B[k][n] = Q[t*16+n][k]
      v2f b = *reinterpret_cast<const v2f*>(
          &lds_q[(t * 16 + m) * QSTRIDE + 4 * c + (g << 1)]);
      d = __builtin_amdgcn_wmma_f32_16x16x4_f32(false, af[c], false, b,
                                                (short)0, d, false, false);
    }
    #pragma unroll
    for (int v = 0; v < 8; ++v)
      oacc[(size_t)v * H_DIM + t * 16] = d[v];
  }
}

extern "C" void kernel_launch(void* const* d_in, const int* in_sizes, int n_in,
                              void* d_out, int out_size, void* d_ws, size_t ws_size,
                              hipStream_t stream) {
  const float* hs = (const float*)d_in[0];   // [8,4096,1024] f32
  const float* Q  = (const float*)d_in[1];   // [1024,16]     f32
  float* out = (float*)d_out;

  const int nTokens = in_sizes[0] / H_DIM;   // 32768
  const int nTiles  = (nTokens + TOK_PER_TILE - 1) / TOK_PER_TILE;
  const int nBlocks = (nTiles + WAVES_PER_BLOCK - 1) / WAVES_PER_BLOCK;

  probe_removal_kernel<<<nBlocks, 256, SMEM_BYTES, stream>>>(hs, Q, out, nTokens);
}